// CPnAction_50525995270866
// MI455X (gfx1250) — compile-verified
//
#include <hip/hip_runtime.h>
#include <hip/hip_bf16.h>
#include <stdint.h>

// ---------------- problem geometry (fixed by the reference) ----------------
#define LSIZE 64
#define SITES (LSIZE * LSIZE)      // 4096 sites per lattice
#define NPHI 6                     // angles per site
#define ROWS_PER_BLK 16            // slab height per block
#define STAGED ((ROWS_PER_BLK + 1) * LSIZE)  // 1088: slab + wrap halo row
#define BLK 256

typedef __attribute__((ext_vector_type(2))) float v2f;
typedef __attribute__((ext_vector_type(8))) float v8f;
typedef __attribute__((ext_vector_type(4))) int   v4i;

#define AS1 __attribute__((address_space(1)))
#define AS3 __attribute__((address_space(3)))

// ============================================================================
// Kernel 1: per-(lattice, 16-row slab) block.
//  Phase A: async-stage phi slab (+halo row) into LDS (16B/lane, ASYNCcnt).
//  Phase B: site map  phi(6) -> z(4 complex) into LDS (sincos + cumprod).
//  Phase C: neighbor dots from LDS, block tree-reduce, 1 partial per block.
// ============================================================================
__global__ __launch_bounds__(BLK) void cpn_site_kernel(
    const float* __restrict__ phi, float* __restrict__ ws)
{
    __shared__ float4 phiS[STAGED * NPHI / 4];  // 1632 * 16B = 26112 B
    __shared__ float4 zS[STAGED * 2];           // 2176 * 16B = 34816 B
    __shared__ float  red[BLK];                 // 1024 B  (total 61952 B)

    const int tid  = threadIdx.x;
    const int bx   = blockIdx.x;   // b * 4 + slab
    const int b    = bx >> 2;
    const int slab = bx & 3;
    const int r0   = slab * ROWS_PER_BLK;

    // ---- Phase A: stage phi rows [r0, r0+16) plus halo row (r0+16)%64 ----
    const size_t mainBase = ((size_t)b * SITES + (size_t)r0 * LSIZE) * NPHI;
    const int    ghalo    = (r0 + ROWS_PER_BLK) & (LSIZE - 1);
    const size_t haloBase = ((size_t)b * SITES + (size_t)ghalo * LSIZE) * NPHI;
    const int MAIN_CH = ROWS_PER_BLK * LSIZE * NPHI / 4;  // 1536 float4 chunks
    const int HALO_CH = LSIZE * NPHI / 4;                 // 96 chunks

    const float4* gm = (const float4*)(phi + mainBase);   // 16B-aligned
    const float4* gh = (const float4*)(phi + haloBase);   // 16B-aligned

#if defined(__gfx1250__) && __has_builtin(__builtin_amdgcn_global_load_async_to_lds_b128)
    for (int c = tid; c < MAIN_CH; c += BLK)
        __builtin_amdgcn_global_load_async_to_lds_b128(
            (v4i*)(gm + c), (AS3 v4i*)(&phiS[c]), 0, 0);
    for (int c = tid; c < HALO_CH; c += BLK)
        __builtin_amdgcn_global_load_async_to_lds_b128(
            (v4i*)(gh + c), (AS3 v4i*)(&phiS[MAIN_CH + c]), 0, 0);
#if __has_builtin(__builtin_amdgcn_s_wait_asynccnt)
    __builtin_amdgcn_s_wait_asynccnt(0);
#else
    asm volatile("s_wait_asynccnt 0" ::: "memory");
#endif
#else
    // fallback: plain vector copies through VGPRs
    for (int c = tid; c < MAIN_CH; c += BLK) phiS[c] = gm[c];
    for (int c = tid; c < HALO_CH; c += BLK) phiS[MAIN_CH + c] = gh[c];
#endif
    __syncthreads();

    // ---- Phase B: spherical map -> z = (zr[4], zi[4]) per staged site ----
    const float* phiF    = (const float*)phiS;
    const float  PI_F    = 3.14159265358979323846f;
    const float  TWO_PI  = 6.28318530717958647692f;
    const float  INV_PI  = 0.318309886183790671538f;
    const float  INV_2PI = 0.159154943091895335769f;

    for (int i = tid; i < STAGED; i += BLK) {
        float x[7];
        float scum = 1.0f;
#pragma unroll
        for (int j = 0; j < NPHI; ++j) {
            float p   = phiF[i * NPHI + j];
            float per = (j == NPHI - 1) ? TWO_PI : PI_F;
            float inv = (j == NPHI - 1) ? INV_2PI : INV_PI;
            float m   = p - floorf(p * inv) * per;  // jnp.mod semantics
            float s, c;
            __sincosf(m, &s, &c);
            x[j] = (j == 0) ? c : c * scum;         // c_j * s_cum[j-1]
            scum *= s;                              // s_cum[j]
        }
        x[6] = scum;                                // s_cum[5]
        zS[i * 2 + 0] = make_float4(x[0], x[1], x[2], x[3]);  // zr
        zS[i * 2 + 1] = make_float4(x[4], x[5], x[6], 0.0f);  // zi (padded)
    }
    __syncthreads();

    // ---- Phase C: link terms sum_d (dre^2 + dim^2 - 1) over 1024 sites ----
    float acc = 0.0f;
#pragma unroll
    for (int k = 0; k < (ROWS_PER_BLK * LSIZE) / BLK; ++k) {  // 4 iterations
        const int i   = tid + k * BLK;
        const int lr  = i >> 6;
        const int col = i & (LSIZE - 1);
        const float4 ur = zS[i * 2 + 0];
        const float4 ui = zS[i * 2 + 1];
        const int n0 = i + LSIZE;                        // roll -1 axis 0 (halo-staged)
        const int n1 = (lr << 6) | ((col + 1) & (LSIZE - 1));  // roll -1 axis 1
#pragma unroll
        for (int d = 0; d < 2; ++d) {
            const int nb = d ? n1 : n0;
            const float4 vr = zS[nb * 2 + 0];
            const float4 vi = zS[nb * 2 + 1];
            float dre = ur.x*vr.x + ur.y*vr.y + ur.z*vr.z + ur.w*vr.w
                      - (ui.x*vi.x + ui.y*vi.y + ui.z*vi.z + ui.w*vi.w);
            float dim = ur.x*vi.x + ur.y*vi.y + ur.z*vi.z + ur.w*vi.w
                      +  ui.x*vr.x + ui.y*vr.y + ui.z*vr.z + ui.w*vr.w;
            acc += dre * dre + dim * dim - 1.0f;
        }
    }

    red[tid] = acc;
    __syncthreads();
    for (int s = BLK / 2; s > 0; s >>= 1) {
        if (tid < s) red[tid] += red[tid + s];
        __syncthreads();
    }
    if (tid == 0) ws[bx] = red[0];
}

// ============================================================================
// Kernel 2: reduce 4 partials per lattice and scale by -N*BETA = -4, via
// V_WMMA_F32_16X16X4_F32: A = 16x4 tile of partials, B = broadcast(-4),
// D[m][0] = -4 * sum_k A[m][k]. One wave handles 16 lattices.
// A layout (32b 16x4): lanes 0-15 -> {K=0 in v0, K=1 in v1}, lanes 16-31 ->
// {K=2, K=3}. D layout: lane 0 holds D[0..7][0], lane 16 holds D[8..15][0].
// ============================================================================
__global__ __launch_bounds__(32) void cpn_reduce_kernel(
    const float* __restrict__ ws, float* __restrict__ out)
{
    const int lane = threadIdx.x;
    const int base = blockIdx.x * 16;
#if defined(__gfx1250__) && __has_builtin(__builtin_amdgcn_wmma_f32_16x16x4_f32)
    const int m  = lane & 15;
    const int k0 = (lane < 16) ? 0 : 2;
    v2f a;
    a.x = ws[(base + m) * 4 + k0 + 0];
    a.y = ws[(base + m) * 4 + k0 + 1];
    v2f bm;
    bm.x = -4.0f;  // -N*BETA, broadcast so B layout is irrelevant
    bm.y = -4.0f;
    v8f c = {};
    v8f d = __builtin_amdgcn_wmma_f32_16x16x4_f32(
        false, a, false, bm, (short)0, c, false, false);
    if (lane == 0 || lane == 16) {
        const int mb = (lane == 0) ? 0 : 8;
#pragma unroll
        for (int v = 0; v < 8; ++v) out[base + mb + v] = d[v];
    }
#else
    if (lane < 16) {
        const float* p = ws + (base + lane) * 4;
        out[base + lane] = -4.0f * (p[0] + p[1] + p[2] + p[3]);
    }
#endif
}

// ============================================================================
extern "C" void kernel_launch(void* const* d_in, const int* in_sizes, int n_in,
                              void* d_out, int out_size, void* d_ws, size_t ws_size,
                              hipStream_t stream)
{
    const float* phi = (const float*)d_in[0];
    // d_in[1] (shift) encodes the fixed roll(-1,axis=0/1) lattice structure,
    // which is exploited analytically via the halo row above.
    (void)in_sizes; (void)n_in; (void)ws_size;

    const int B = in_sizes[0] / (SITES * NPHI);  // 1024
    float* ws  = (float*)d_ws;                   // B*4 partial sums
    float* out = (float*)d_out;                  // B floats

    cpn_site_kernel<<<dim3(B * 4), dim3(BLK), 0, stream>>>(phi, ws);
    cpn_reduce_kernel<<<dim3(B / 16), dim3(32), 0, stream>>>(ws, out);
}